// GraphConvolution_11493332484390
// MI455X (gfx1250) — compile-verified
//
#include <hip/hip_runtime.h>

typedef __attribute__((ext_vector_type(2))) float v2f;
typedef __attribute__((ext_vector_type(8))) float v8f;

#define NNODES 100000
#define INF    256
#define OUTF   128
#define LDS_PITCH 260   // 256 + 4 pad: row-to-row bank offset of 4 -> conflict-free col reads

// ---------------------------------------------------------------------------
// out[n][f] = bias[f]   (vectorized float4; N*OUT/4 = 3.2M float4s)
// ---------------------------------------------------------------------------
__global__ __launch_bounds__(256) void gcn_init_kernel(const float* __restrict__ bias,
                                                       float* __restrict__ out) {
    int i = blockIdx.x * blockDim.x + threadIdx.x;     // float4 index
    float4 b = ((const float4*)bias)[i & (OUTF / 4 - 1)];
    ((float4*)out)[i] = b;
}

// ---------------------------------------------------------------------------
// support = x @ W  via V_WMMA_F32_16X16X4_F32
// block = 256 threads (8 waves). Block handles 16 rows x 128 cols.
// Wave w owns the 16x16 tile at cols [16w, 16w+16).
// ---------------------------------------------------------------------------
__global__ __launch_bounds__(256) void gcn_gemm_kernel(const float* __restrict__ x,
                                                       const float* __restrict__ w,
                                                       float* __restrict__ support) {
    __shared__ float lds_x[16 * LDS_PITCH];

    const int t    = threadIdx.x;
    const int row0 = blockIdx.x * 16;

    // Stage x[row0 .. row0+16)[0..256) into LDS, float4-coalesced.
    const float4* xg = (const float4*)(x + (long long)row0 * INF);
    #pragma unroll
    for (int i = t; i < 16 * (INF / 4); i += 256) {
        int r  = i >> 6;          // row 0..15
        int c4 = i & 63;          // float4 col 0..63
        float4 v = xg[r * (INF / 4) + c4];
        float* dst = &lds_x[r * LDS_PITCH + c4 * 4];
        dst[0] = v.x; dst[1] = v.y; dst[2] = v.z; dst[3] = v.w;
    }
    __syncthreads();

    const int wave = t >> 5;
    const int lane = t & 31;
    const int lrow = lane & 15;
    const int hi   = lane >> 4;          // 0 for lanes 0-15, 1 for lanes 16-31
    const int n    = wave * 16 + lrow;   // output column owned by this lane
    const int ksel = hi * 2;             // hi half-wave carries K+2, K+3 slices

    const float* lx = &lds_x[lrow * LDS_PITCH];

    v8f acc = {};
    #pragma unroll 8
    for (int k0 = 0; k0 < INF; k0 += 4) {
        // A (16x4 f32): lanes 0-15 -> {K0,K1}, lanes 16-31 -> {K2,K3}, row = lane%16
        v2f a;
        a.x = lx[k0 + ksel + 0];
        a.y = lx[k0 + ksel + 1];
        // B (4x16 f32): vgpr0 -> K0(lo)/K2(hi), vgpr1 -> K1(lo)/K3(hi), col = n
        v2f b;
        b.x = w[(k0 + ksel + 0) * OUTF + n];
        b.y = w[(k0 + ksel + 1) * OUTF + n];
        acc = __builtin_amdgcn_wmma_f32_16x16x4_f32(
            /*neg_a=*/false, a, /*neg_b=*/false, b,
            /*c_mod=*/(short)0, acc, /*reuse_a=*/false, /*reuse_b=*/false);
    }

    // C/D layout: vgpr g, lanes 0-15 -> M=g, lanes 16-31 -> M=g+8; N = lane%16 tile col
    const int mbase = row0 + hi * 8;
    #pragma unroll
    for (int g = 0; g < 8; ++g) {
        support[(long long)(mbase + g) * OUTF + n] = acc[g];
    }
}

// ---------------------------------------------------------------------------
// SpMM scatter: one wave per edge. out[row[e]][:] += vals[e] * support[col[e]][:]
// support (51 MB) is L2-resident; atomics land in L2.
// ---------------------------------------------------------------------------
__global__ __launch_bounds__(256) void gcn_spmm_kernel(const float* __restrict__ support,
                                                       const float* __restrict__ vals,
                                                       const int* __restrict__ rowi,
                                                       const int* __restrict__ coli,
                                                       float* __restrict__ out,
                                                       int nedges) {
    int e = blockIdx.x * (blockDim.x >> 5) + (threadIdx.x >> 5);
    if (e >= nedges) return;
    int lane = threadIdx.x & 31;

    int   r = rowi[e];
    int   c = coli[e];
    float v = vals[e];

    float4 s = ((const float4*)(support + (long long)c * OUTF))[lane];
    float* o = out + (long long)r * OUTF + lane * 4;
    atomicAdd(o + 0, v * s.x);
    atomicAdd(o + 1, v * s.y);
    atomicAdd(o + 2, v * s.z);
    atomicAdd(o + 3, v * s.w);
}

// ---------------------------------------------------------------------------
extern "C" void kernel_launch(void* const* d_in, const int* in_sizes, int n_in,
                              void* d_out, int out_size, void* d_ws, size_t ws_size,
                              hipStream_t stream) {
    const float* x    = (const float*)d_in[0];
    const float* vals = (const float*)d_in[1];
    const float* w    = (const float*)d_in[2];
    const float* bias = (const float*)d_in[3];
    const int*   rowi = (const int*)d_in[4];
    const int*   coli = (const int*)d_in[5];

    float* out     = (float*)d_out;
    float* support = (float*)d_ws;          // N*OUT floats = 51.2 MB scratch

    const int nnodes = in_sizes[0] / INF;   // 100000
    const int nedges = in_sizes[4];         // 3200000

    // 1) out = broadcast(bias)
    {
        int total4 = nnodes * (OUTF / 4);
        gcn_init_kernel<<<(total4 + 255) / 256, 256, 0, stream>>>(bias, out);
    }
    // 2) support = x @ W   (fp32 WMMA)
    {
        int nblocks = (nnodes + 15) / 16;   // 6250
        gcn_gemm_kernel<<<nblocks, 256, 0, stream>>>(x, w, support);
    }
    // 3) out += spmm(adj, support)
    {
        int waves_per_block = 256 / 32;     // 8 edges per block
        int nblocks = (nedges + waves_per_block - 1) / waves_per_block;
        gcn_spmm_kernel<<<nblocks, 256, 0, stream>>>(support, vals, rowi, coli, out, nedges);
    }
}